// idxLayer2_58514634441009
// MI455X (gfx1250) — compile-verified
//
#include <hip/hip_runtime.h>
#include <stdint.h>

// Gather-and-pack:  out[h] = concat( x[idx[h,0..31]] (32x64 f32), dis[h], sin[h], cos[h] )
// Pure data movement -> CDNA5 async global<->LDS DMA path, with temporal hints:
//   - x gather loads: RT (keep the 25.6 MB table resident in the 192 MB L2)
//   - output stores + single-use streams: NT (don't thrash L2)

#define WROW 32                       // neighbors per row
#define DDIM 64                       // feature dim
#define GATHER_BYTES (WROW * DDIM * 4)           // 8192 B of gathered x per row
#define TAIL_BYTES   (3 * WROW * 4)              // 384 B (dis/sin/cos)
#define ROW_BYTES    (GATHER_BYTES + TAIL_BYTES) // 8576 B per output row

__global__ __launch_bounds__(256) void gather_pack_async(
    const float* __restrict__ x,
    const long long* __restrict__ idx,
    const float* __restrict__ dis,
    const float* __restrict__ sn,
    const float* __restrict__ cs,
    float* __restrict__ out,
    int H)
{
    __shared__ __align__(16) unsigned char smem[ROW_BYTES];

    const int h = blockIdx.x;
    if (h >= H) return;
    const int t   = threadIdx.x;   // 0..255
    const int w   = t >> 3;        // neighbor index 0..31
    const int sub = t & 7;         // 32-byte chunk within the 256-byte feature row

    // ---------------- stage 1: async gather x[idx[h,w]] -> LDS ----------------
    // idx is read exactly once per dispatch -> non-temporal.
    const long long nbr = __builtin_nontemporal_load(&idx[(size_t)h * WROW + w]);
    const unsigned long long src =
        (unsigned long long)(uintptr_t)x +
        ((unsigned long long)nbr << 8) +            // nbr * 256 B
        ((unsigned)sub << 5);                       // sub * 32 B
    // low 32 bits of the generic pointer to a __shared__ object == LDS byte address
    const unsigned int lds0 = (unsigned int)(uintptr_t)&smem[t * 32];

    // IOFFSET is added to BOTH the global and the LDS address (ISA 08 / 4.4),
    // so offset:16 moves the second 16B half with the same base operands.
    // Default (RT) temporal hint: keep x L2-resident across its ~32x reuse.
    asm volatile("global_load_async_to_lds_b128 %0, %1, off"
                 :: "v"(lds0), "v"(src) : "memory");
    asm volatile("global_load_async_to_lds_b128 %0, %1, off offset:16"
                 :: "v"(lds0), "v"(src) : "memory");

    // ---------------- tail: dis/sin/cos (384 B = 24 x b128) --------------------
    unsigned int       ldsT = 0;
    unsigned long long dstT = 0;
    if (t < 24) {
        const int grp   = t >> 3;   // 0=dis, 1=sin, 2=cos
        const int lane8 = t & 7;
        const float* base = (grp == 0) ? dis : (grp == 1) ? sn : cs;
        const unsigned long long tsrc =
            (unsigned long long)(uintptr_t)base +
            (unsigned long long)h * (WROW * 4) +    // 128 B per row, 16B aligned
            (unsigned)lane8 * 16;
        ldsT = (unsigned int)(uintptr_t)&smem[GATHER_BYTES + t * 16];
        // Single-use stream -> non-temporal load.
        asm volatile("global_load_async_to_lds_b128 %0, %1, off th:TH_LOAD_NT"
                     :: "v"(ldsT), "v"(tsrc) : "memory");
        dstT = (unsigned long long)(uintptr_t)out +
               (unsigned long long)h * ROW_BYTES + GATHER_BYTES + (unsigned)t * 16;
    }

    // Each thread stores only bytes it loaded itself -> per-wave wait suffices,
    // no workgroup barrier required (ASYNCcnt is per-wave).
    asm volatile("s_wait_asynccnt 0x0" ::: "memory");

    // ---------------- stage 2: async store LDS -> out (fully coalesced) -------
    // Output is written once and never re-read -> non-temporal stores so the
    // 858 MB write stream does not evict the x table from L2.
    const unsigned long long dst =
        (unsigned long long)(uintptr_t)out +
        (unsigned long long)h * ROW_BYTES + (unsigned)t * 32;
    asm volatile("global_store_async_from_lds_b128 %0, %1, off th:TH_STORE_NT"
                 :: "v"(dst), "v"(lds0) : "memory");
    asm volatile("global_store_async_from_lds_b128 %0, %1, off offset:16 th:TH_STORE_NT"
                 :: "v"(dst), "v"(lds0) : "memory");
    if (t < 24) {
        asm volatile("global_store_async_from_lds_b128 %0, %1, off th:TH_STORE_NT"
                     :: "v"(dstT), "v"(ldsT) : "memory");
    }

    // Drain async stores before the wave retires.
    asm volatile("s_wait_asynccnt 0x0" ::: "memory");
}

extern "C" void kernel_launch(void* const* d_in, const int* in_sizes, int n_in,
                              void* d_out, int out_size, void* d_ws, size_t ws_size,
                              hipStream_t stream) {
    const float*     x   = (const float*)d_in[0];
    const long long* idx = (const long long*)d_in[1];   // jnp.int64
    const float*     dis = (const float*)d_in[2];
    const float*     sn  = (const float*)d_in[3];
    const float*     cs  = (const float*)d_in[4];
    float* out = (float*)d_out;

    const int H = in_sizes[2] / WROW;   // dis has H*W elements
    gather_pack_async<<<H, 256, 0, stream>>>(x, idx, dis, sn, cs, out, H);

    (void)n_in; (void)out_size; (void)d_ws; (void)ws_size;
}